// GCNet_66984309948603
// MI455X (gfx1250) — compile-verified
//
#include <hip/hip_runtime.h>

// MI455X / gfx1250 implementation.
// Pipeline:
//   1) k_edge_agg   : aggx[N,5] += x[src], deg[dst] += 1   (atomics)
//   2) k_node_h     : h = relu((aggx/deg)@W1 + x@root1 + b1)   [relu∘leaky == relu]
//   3) k_edge_asc   : A_sc[src,:] += attr * pos[dst,:]     (wave per edge, f32 atomics)
//   4) k_pool_wmma  : xp = pos^T @ h, adj = pos^T @ A_sc   via V_WMMA_F32_16X16X4_F32
//   5) k_finish     : mask/deg2/agg2/h2/global-mean/MLP -> out[2]
// Precision: everything fp32 (xp feeds the output directly; fp32 WMMA keeps
// reference-level accuracy and the pipeline is L2/atomics bound anyway).

typedef __attribute__((ext_vector_type(2))) float v2f;
typedef __attribute__((ext_vector_type(4))) float v4f;
typedef __attribute__((ext_vector_type(8))) float v8f;

#define INC    5
#define H1C    16
#define H2C    32
#define KCL    256      // clusters
#define KSPLIT 25       // K-dim split for the pooling GEMMs
#define MTILES 16       // 256/16

// ---------------------------------------------------------------- kernel 1
__global__ void k_edge_agg(const float* __restrict__ x, const int* __restrict__ ei,
                           float* __restrict__ aggx, float* __restrict__ deg, int E) {
    int e = blockIdx.x * blockDim.x + threadIdx.x;
    if (e >= E) return;
    int s = ei[e];
    int d = ei[E + e];
#pragma unroll
    for (int c = 0; c < INC; ++c)
        atomicAdd(&aggx[(size_t)d * INC + c], x[(size_t)s * INC + c]);
    atomicAdd(&deg[d], 1.0f);
}

// ---------------------------------------------------------------- kernel 2
__global__ void k_node_h(const float* __restrict__ x, const float* __restrict__ aggx,
                         const float* __restrict__ deg, const float* __restrict__ W1,
                         const float* __restrict__ root1, const float* __restrict__ b1,
                         float* __restrict__ h, int N) {
    __shared__ float sW[INC * H1C], sR[INC * H1C], sb[H1C];
    int t = threadIdx.x;
    if (t < INC * H1C) { sW[t] = W1[t]; sR[t] = root1[t]; }
    if (t < H1C) sb[t] = b1[t];
    __syncthreads();
    int i = blockIdx.x * blockDim.x + t;
    if (i >= N) return;
    float dn = fmaxf(deg[i], 1.0f);
    float xi[INC], ai[INC];
#pragma unroll
    for (int c = 0; c < INC; ++c) {
        xi[c] = x[(size_t)i * INC + c];
        ai[c] = aggx[(size_t)i * INC + c] / dn;
    }
#pragma unroll
    for (int c = 0; c < H1C; ++c) {
        float v = sb[c];
#pragma unroll
        for (int q = 0; q < INC; ++q)
            v += ai[q] * sW[q * H1C + c] + xi[q] * sR[q * H1C + c];
        h[(size_t)i * H1C + c] = fmaxf(v, 0.0f);   // leaky(relu(v)) == relu(v)
    }
}

// ---------------------------------------------------------------- kernel 3
// one wave per edge; lane handles 8 contiguous columns (two float4 loads)
__global__ void k_edge_asc(const float* __restrict__ pos, const int* __restrict__ ei,
                           const float* __restrict__ attr, float* __restrict__ Asc, int E) {
    int lane = threadIdx.x & 31;
    int e = blockIdx.x * (blockDim.x >> 5) + (threadIdx.x >> 5);
    if (e >= E) return;
    int s = ei[e];
    int d = ei[E + e];
    float a = attr[e];
    const float* prow = pos + (size_t)d * KCL + lane * 8;
    float*       arow = Asc + (size_t)s * KCL + lane * 8;
    v4f p0 = *(const v4f*)(prow);
    v4f p1 = *(const v4f*)(prow + 4);
#pragma unroll
    for (int j = 0; j < 4; ++j) atomicAdd(&arow[j],     a * p0[j]);
#pragma unroll
    for (int j = 0; j < 4; ++j) atomicAdd(&arow[4 + j], a * p1[j]);
}

// ---------------------------------------------------------------- kernel 4
// K-loop specialized on the B leading dimension so all intra-iteration
// offsets are compile-time immediates; only two 64-bit pointer bumps per
// 4-WMMA iteration remain.
template <int LD>
__device__ __forceinline__ void pool_mac_loop(
    const float* __restrict__ Ab, const float* __restrict__ Bb,
    int row0, int row1, int N, int hi, v8f& acc) {
    const float* pa = Ab + (size_t)(row0 + 2 * hi) * KCL;
    const float* pb = Bb + (size_t)(row0 + 2 * hi) * LD;
    int r = row0;
    // bulk: 16 rows / 4 WMMAs per iteration, constant-immediate offsets
    for (; r + 16 <= row1; r += 16) {
#pragma unroll
        for (int u = 0; u < 4; ++u) {
            v2f A, B;
            A.x = pa[(u * 4 + 0) * KCL];
            A.y = pa[(u * 4 + 1) * KCL];
            B.x = pb[(u * 4 + 0) * LD];
            B.y = pb[(u * 4 + 1) * LD];
            acc = __builtin_amdgcn_wmma_f32_16x16x4_f32(
                false, A, false, B, (short)0, acc, false, false);
        }
        pa += 16 * KCL;
        pb += 16 * LD;
    }
    // remaining full quads (0..3)
    for (; r + 4 <= row1; r += 4) {
        v2f A, B;
        A.x = pa[0];
        A.y = pa[KCL];
        B.x = pb[0];
        B.y = pb[LD];
        acc = __builtin_amdgcn_wmma_f32_16x16x4_f32(
            false, A, false, B, (short)0, acc, false, false);
        pa += 4 * KCL;
        pb += 4 * LD;
    }
    // partial quad: VALU-select masking (EXEC must stay all-1s for WMMA)
    if (r < row1) {
        int ra = r + 2 * hi;
        int rb = ra + 1;
        float ma = (ra < row1) ? 1.0f : 0.0f;
        float mb = (rb < row1) ? 1.0f : 0.0f;
        int rca = (ra < N - 1) ? ra : (N - 1);
        int rcb = (rb < N - 1) ? rb : (N - 1);
        v2f A, B;
        A.x = Ab[(size_t)rca * KCL] * ma;
        A.y = Ab[(size_t)rcb * KCL] * mb;
        B.x = Bb[(size_t)rca * LD] * ma;
        B.y = Bb[(size_t)rcb * LD] * mb;
        acc = __builtin_amdgcn_wmma_f32_16x16x4_f32(
            false, A, false, B, (short)0, acc, false, false);
    }
}

// block = 17 waves (544 thr): wave w<16 -> adj n-tile w, wave 16 -> xp tile.
// blockIdx.x encodes (m-tile, k-chunk). All waves share the same pos m-slice
// (WGP$ reuse across the 17 waves). fp32 WMMA 16x16x4:
//   A (16x4, MxK)  lane L: M=L&15; VGPR0=K(2*hi), VGPR1=K(2*hi+1), hi=L>>4
//   B (4x16, KxN)  lane L: N=L&15; VGPR0=K(2*hi), VGPR1=K(2*hi+1)
//   D (16x16)      VGPR r at lane L = C[row r+8*hi, col L&15]
__global__ void __launch_bounds__(544) k_pool_wmma(
    const float* __restrict__ pos, const float* __restrict__ h,
    const float* __restrict__ Asc, float* __restrict__ xp,
    float* __restrict__ adj, int N) {
    int wave = threadIdx.x >> 5;         // 0..16
    int lane = threadIdx.x & 31;
    int mt = blockIdx.x / KSPLIT;
    int kc = blockIdx.x % KSPLIT;
    int chunk = (((N + KSPLIT - 1) / KSPLIT) + 3) & ~3;   // multiple of 4
    int row0 = kc * chunk;
    int row1 = row0 + chunk;
    if (row1 > N) row1 = N;
    if (row0 >= row1) return;            // uniform per block

    int M  = lane & 15;
    int hi = lane >> 4;

    const float* Ab = pos + mt * 16 + M;            // column mt*16+M of pos

    v8f acc = {0.f, 0.f, 0.f, 0.f, 0.f, 0.f, 0.f, 0.f};

    if (wave < 16) {
        pool_mac_loop<KCL>(Ab, Asc + wave * 16 + M, row0, row1, N, hi, acc);
        float* dst = adj + (size_t)(mt * 16 + hi * 8) * KCL + wave * 16 + M;
#pragma unroll
        for (int rr = 0; rr < 8; ++rr) atomicAdd(dst + (size_t)rr * KCL, acc[rr]);
    } else {
        pool_mac_loop<H1C>(Ab, h + M, row0, row1, N, hi, acc);
        float* dst = xp + (size_t)(mt * 16 + hi * 8) * H1C + M;
#pragma unroll
        for (int rr = 0; rr < 8; ++rr) atomicAdd(dst + (size_t)rr * H1C, acc[rr]);
    }
}

// ---------------------------------------------------------------- kernel 5
__global__ void __launch_bounds__(256) k_finish(
    const float* __restrict__ xp, const float* __restrict__ adj,
    const float* __restrict__ W2, const float* __restrict__ root2,
    const float* __restrict__ b2,
    const float* __restrict__ lw1, const float* __restrict__ lb1,
    const float* __restrict__ lw2, const float* __restrict__ lb2,
    float* __restrict__ out) {
    __shared__ float sW2[H1C * H2C];
    __shared__ float sR2[H1C * H2C];
    __shared__ float sb2v[H2C];
    __shared__ float sxp[KCL * H1C];
    __shared__ float sxw[KCL * H2C];   // xw, later reused as h2
    __shared__ float sg[H2C];
    int t = threadIdx.x;
    for (int i = t; i < H1C * H2C; i += 256) { sW2[i] = W2[i]; sR2[i] = root2[i]; }
    if (t < H2C) sb2v[t] = b2[t];
    for (int i = t; i < KCL * H1C; i += 256) sxp[i] = xp[i];
    __syncthreads();

    // xw = xp @ W2   (thread t == cluster row)
    {
        float xk[H1C];
#pragma unroll
        for (int q = 0; q < H1C; ++q) xk[q] = sxp[t * H1C + q];
#pragma unroll
        for (int c = 0; c < H2C; ++c) {
            float v = 0.0f;
#pragma unroll
            for (int q = 0; q < H1C; ++q) v += xk[q] * sW2[q * H2C + c];
            sxw[t * H2C + c] = v;
        }
    }
    __syncthreads();

    // deg2 / agg2 for target column n = t (coalesced column reads of adj)
    float acc[H2C];
#pragma unroll
    for (int c = 0; c < H2C; ++c) acc[c] = 0.0f;
    float d2 = 0.0f;
    for (int m = 0; m < KCL; ++m) {
        float av = adj[(size_t)m * KCL + t];
        if (av != 0.0f) {
            d2 += 1.0f;
#pragma unroll
            for (int c = 0; c < H2C; ++c) acc[c] += sxw[m * H2C + c];
        }
    }
    float dinv = 1.0f / fmaxf(d2, 1.0f);
    float h2v[H2C];
#pragma unroll
    for (int c = 0; c < H2C; ++c) {
        float rv = sb2v[c];
#pragma unroll
        for (int q = 0; q < H1C; ++q) rv += sxp[t * H1C + q] * sR2[q * H2C + c];
        h2v[c] = fmaxf(acc[c] * dinv + rv, 0.0f);   // leaky(relu) == relu
    }
    __syncthreads();                  // done reading sxw as xw
#pragma unroll
    for (int c = 0; c < H2C; ++c) sxw[t * H2C + c] = h2v[c];   // now h2
    __syncthreads();

    if (t < H2C) {
        float s = 0.0f;
        for (int m = 0; m < KCL; ++m) s += sxw[m * H2C + t];
        sg[t] = s * (1.0f / KCL);
    }
    __syncthreads();

    if (t == 0) {
        float tmp[8];
        for (int j = 0; j < 8; ++j) {
            float v = lb1[j];
            for (int c = 0; c < H2C; ++c) v += sg[c] * lw1[c * 8 + j];
            tmp[j] = (v > 0.0f) ? v : 0.1f * v;    // genuine leaky here
        }
        for (int i = 0; i < 2; ++i) {
            float v = lb2[i];
            for (int j = 0; j < 8; ++j) v += tmp[j] * lw2[j * 2 + i];
            out[i] = v;
        }
    }
}

// ---------------------------------------------------------------- launch
extern "C" void kernel_launch(void* const* d_in, const int* in_sizes, int n_in,
                              void* d_out, int out_size, void* d_ws, size_t ws_size,
                              hipStream_t stream) {
    const float* x     = (const float*)d_in[0];
    const int*   ei    = (const int*)d_in[1];     // [2,E] (src row, dst row)
    const float* attr  = (const float*)d_in[2];
    const float* pos   = (const float*)d_in[3];
    const float* W1    = (const float*)d_in[4];
    const float* root1 = (const float*)d_in[5];
    const float* b1    = (const float*)d_in[6];
    const float* W2    = (const float*)d_in[7];
    const float* root2 = (const float*)d_in[8];
    const float* b2    = (const float*)d_in[9];
    const float* lw1   = (const float*)d_in[10];
    const float* lb1   = (const float*)d_in[11];
    const float* lw2   = (const float*)d_in[12];
    const float* lb2   = (const float*)d_in[13];

    int N = in_sizes[0] / INC;
    int E = in_sizes[2];

    float* ws   = (float*)d_ws;
    float* aggx = ws;                              // N*5
    float* deg  = aggx + (size_t)N * INC;          // N
    float* h    = deg + (size_t)N;                 // N*16
    float* Asc  = h + (size_t)N * H1C;             // N*256
    float* xp   = Asc + (size_t)N * KCL;           // 256*16
    float* adj  = xp + (size_t)KCL * H1C;          // 256*256

    hipMemsetAsync(aggx, 0, (size_t)N * (INC + 1) * sizeof(float), stream);
    hipMemsetAsync(Asc,  0, (size_t)N * KCL * sizeof(float), stream);
    hipMemsetAsync(xp,   0, (size_t)(KCL * H1C + KCL * KCL) * sizeof(float), stream);

    k_edge_agg<<<(E + 255) / 256, 256, 0, stream>>>(x, ei, aggx, deg, E);
    k_node_h<<<(N + 255) / 256, 256, 0, stream>>>(x, aggx, deg, W1, root1, b1, h, N);
    k_edge_asc<<<(E + 7) / 8, 256, 0, stream>>>(pos, ei, attr, Asc, E);
    k_pool_wmma<<<MTILES * KSPLIT, 544, 0, stream>>>(pos, h, Asc, xp, adj, N);
    k_finish<<<1, 256, 0, stream>>>(xp, adj, W2, root2, b2, lw1, lb1, lw2, lb2,
                                    (float*)d_out);
}